// MultiHeadAttention_24739011624998
// MI455X (gfx1250) — compile-verified
//
#include <hip/hip_runtime.h>
#include <hip/hip_bf16.h>

// MI455X / gfx1250, wave32. D = A(16x4 f32) * B(4x16 f32) + C(16x16 f32)
typedef float v2f __attribute__((ext_vector_type(2)));
typedef float v8f __attribute__((ext_vector_type(8)));

#define WMMA_F32(acc, a, b) \
  __builtin_amdgcn_wmma_f32_16x16x4_f32(false, (a), false, (b), (short)0, (acc), false, false)

// Problem constants
constexpr int Bb  = 2;
constexpr int Ss  = 2048;
constexpr int Dd  = 512;
constexpr int Hh  = 8;
constexpr int DH  = 64;
constexpr int DSP = 16;
constexpr int MROWS = Bb * Ss;                            // 4096
constexpr size_t PROJ_ELEMS = (size_t)Bb * Hh * Ss * DH;  // 2,097,152

// ---------------------------------------------------------------------------
// CDNA5 async global->LDS (ASYNCcnt-tracked, bypasses VGPRs).
// VDST = per-lane LDS byte address, VADDR = 64-bit global address.
// Low 32 bits of a generic pointer to __shared__ are the LDS byte address.
// ---------------------------------------------------------------------------
__device__ __forceinline__ void async_g2l_b128(void* lds, const void* g) {
  const unsigned lds_off = (unsigned)(size_t)lds;
  asm volatile("global_load_async_to_lds_b128 %0, %1, off"
               :: "v"(lds_off), "v"(g) : "memory");
}
__device__ __forceinline__ void wait_async0() {
  asm volatile("s_wait_asynccnt 0x0" ::: "memory");
}

// ---------------------------------------------------------------------------
// Kernel 1: GEMM  dst = X[4096,512] * W[512,512] + bias, WMMA f32.
// Double-buffered LDS, staged with async b128 copies overlapped with WMMA.
// dst_head_layout=1 -> scatter into [B,H,S,DH]; =0 -> row-major [B,S,D].
// Block: 256 threads (8 waves), tile 32(M) x 64(N), K-blocked by 16.
// ---------------------------------------------------------------------------
__global__ __launch_bounds__(256) void proj_gemm(
    const float* __restrict__ X, const float* __restrict__ W,
    const float* __restrict__ bias, float* __restrict__ dst,
    int dst_head_layout) {
  // As stride 20 floats: rows 80B apart -> b128 stores aligned, v2f reads aligned
  __shared__ float As[2][32][20];
  __shared__ float Bs[2][16][64];

  const int t    = threadIdx.x;
  const int w    = t >> 5;
  const int lane = t & 31;
  const int hb   = lane >> 4;    // half-wave: selects K pair within frag
  const int ln   = lane & 15;
  const int wrow = (w & 1) * 16; // 2x4 grid of wave tiles -> 32x64 block tile
  const int wcol = (w >> 1) * 16;
  const int m0   = blockIdx.x * 32;
  const int n0   = blockIdx.y * 64;

  // stage K-block k0 into LDS buffer buf (async, 16B per thread)
  auto issue = [&](int buf, int k0) {
    if (t < 128) {                         // A: 32x16 floats = 128 x 16B
      const int row = t >> 2, col = (t & 3) * 4;
      async_g2l_b128(&As[buf][row][col],
                     X + (size_t)(m0 + row) * Dd + k0 + col);
    }
    {                                      // B: 16x64 floats = 256 x 16B
      const int row = t >> 4, col = (t & 15) * 4;
      async_g2l_b128(&Bs[buf][row][col],
                     W + (size_t)(k0 + row) * Dd + n0 + col);
    }
  };

  issue(0, 0);
  v8f acc = {};
  for (int kb = 0; kb < Dd / 16; ++kb) {
    wait_async0();        // my chunks issued last iter have landed
    __syncthreads();      // everyone waited -> whole buffer visible
    if (kb + 1 < Dd / 16) issue((kb + 1) & 1, (kb + 1) * 16);  // overlap w/ WMMA
    const int cur = kb & 1;
#pragma unroll
    for (int s = 0; s < 4; ++s) {
      const int kl = s * 4 + hb * 2;
      v2f a = *(const v2f*)&As[cur][wrow + ln][kl];  // A[m][k], m=ln
      v2f b;                                          // B[k][n], n=ln
      b.x = Bs[cur][kl][wcol + ln];
      b.y = Bs[cur][kl + 1][wcol + ln];
      acc = WMMA_F32(acc, a, b);
    }
  }
  // epilogue: C/D layout -> vgpr r: lanes0-15 M=r, lanes16-31 M=r+8; N=ln
#pragma unroll
  for (int r = 0; r < 8; ++r) {
    const int mrow = m0 + wrow + r + hb * 8;
    const int ncol = n0 + wcol + ln;
    const float v = acc[r] + bias[ncol];
    if (dst_head_layout) {
      const int b = mrow >> 11, s = mrow & (Ss - 1);
      const int h = ncol >> 6, dh = ncol & (DH - 1);
      dst[(((size_t)b * Hh + h) * Ss + s) * DH + dh] = v;
    } else {
      dst[(size_t)mrow * Dd + ncol] = v;
    }
  }
}

// ---------------------------------------------------------------------------
// Kernel 2: fused attention for one (b,h, 16-query tile).
//  - QK^T via WMMA f32 (Q frags resident in VGPRs)
//  - fused spatial-MLP bias, logits -> 128KB LDS
//  - rowwise softmax (shfl_xor wave reductions), attn streamed to d_out
//  - AV via WMMA f32 with A read from LDS attn
// Block: 256 threads (8 waves). Grid: B*H*(S/16) = 2048.
// ---------------------------------------------------------------------------
__global__ __launch_bounds__(256) void attention_kernel(
    const float* __restrict__ qh, const float* __restrict__ kh,
    const float* __restrict__ vh, const float* __restrict__ mind,
    const float* __restrict__ sw1, const float* __restrict__ sb1,
    const float* __restrict__ sw2, const float* __restrict__ sb2,
    float* __restrict__ attn_out, float* __restrict__ oh) {
  __shared__ float logits[16][Ss + 2];   // stride 2050 (even): aligned + conflict-free
  __shared__ float avbuf[16][64];
  __shared__ float spw[3 * DSP + 1];     // sw1 | sb1 | sw2 | sb2

  const int t    = threadIdx.x;
  const int w    = t >> 5;
  const int lane = t & 31;
  const int hb   = lane >> 4;
  const int ln   = lane & 15;
  const int qt   = blockIdx.x & 127;     // S/16 = 128
  const int bh   = blockIdx.x >> 7;      // b*H + h
  const int b    = bh >> 3;
  const int q0   = qt * 16;

  if (t < DSP) { spw[t] = sw1[t]; spw[DSP + t] = sb1[t]; spw[2 * DSP + t] = sw2[t]; }
  if (t == DSP) spw[3 * DSP] = sb2[0];

  const float* qbase = qh + (size_t)bh * Ss * DH;
  const float* kbase = kh + (size_t)bh * Ss * DH;
  const float* vbase = vh + (size_t)bh * Ss * DH;

  // Q A-fragments for the whole tile: A[m][k], m=ln, K = 0..63
  v2f aq[16];
#pragma unroll
  for (int kk = 0; kk < 16; ++kk) {
    const int d0 = kk * 4 + hb * 2;
    aq[kk] = *(const v2f*)(qbase + (size_t)(q0 + ln) * DH + d0);
  }
  __syncthreads();  // spw visible

  const float scale = 0.125f;  // 1/sqrt(64)

  // ---- QK^T + spatial bias: each wave owns a 16-wide K stripe per iter ----
  for (int it = 0; it < 16; ++it) {
    const int kb = it * 128 + w * 16;
    if (it + 1 < 16)  // pull next K-stripe toward the caches (global_prefetch_b8)
      __builtin_prefetch(kbase + (size_t)(kb + 128 + lane / 2) * DH + (lane & 1) * 32, 0, 0);
    v8f acc = {};
#pragma unroll
    for (int kk = 0; kk < 16; ++kk) {
      const int d0 = kk * 4 + hb * 2;
      // B[d][n] = kh[kb+n][d], n=ln : contiguous in d -> v2f load
      v2f bf = *(const v2f*)(kbase + (size_t)(kb + ln) * DH + d0);
      acc = WMMA_F32(acc, aq[kk], bf);
    }
#pragma unroll
    for (int r = 0; r < 8; ++r) {
      const int ql = r + hb * 8;
      const int kc = kb + ln;
      const float m = mind[((size_t)b * Ss + q0 + ql) * Ss + kc];
      float s = spw[3 * DSP];
#pragma unroll
      for (int j = 0; j < DSP; ++j) {
        const float hj = fmaxf(fmaf(m, spw[j], spw[DSP + j]), 0.0f);
        s = fmaf(hj, spw[2 * DSP + j], s);
      }
      logits[ql][kc] = acc[r] * scale + fmaxf(s, 0.0f);
    }
  }
  __syncthreads();

  // ---- softmax: wave w handles rows 2w, 2w+1 ----
#pragma unroll
  for (int rr = 0; rr < 2; ++rr) {
    const int ql = w * 2 + rr;
    float mx = -3.4e38f;
    for (int c = lane; c < Ss; c += 32) mx = fmaxf(mx, logits[ql][c]);
#pragma unroll
    for (int off = 16; off > 0; off >>= 1) mx = fmaxf(mx, __shfl_xor(mx, off, 32));
    float sum = 0.0f;
    for (int c = lane; c < Ss; c += 32) {
      const float e = __expf(logits[ql][c] - mx);
      logits[ql][c] = e;
      sum += e;
    }
#pragma unroll
    for (int off = 16; off > 0; off >>= 1) sum += __shfl_xor(sum, off, 32);
    const float inv = 1.0f / sum;
    float* arow = attn_out + ((size_t)bh * Ss + q0 + ql) * Ss;
    for (int c = lane; c < Ss; c += 32) {
      const float a = logits[ql][c] * inv;
      logits[ql][c] = a;     // keep normalized attn for AV
      arow[c] = a;           // materialize attn output (coalesced)
    }
  }
  __syncthreads();

  // ---- AV: out[16][64] = attn[16][2048] * vh[2048][64] ----
  // wave -> (n-tile = w&3, K-half = w>>2); halves combined via LDS.
  const int nt = (w & 3) * 16;
  const int half = w >> 2;
  v8f oacc = {};
  for (int ks = 0; ks < 256; ++ks) {
    const int kl = half * 1024 + ks * 4 + hb * 2;
    v2f a = *(const v2f*)&logits[ln][kl];             // A[m][k], m=ln
    v2f bf;                                            // B[k][n] = vh[kl][nt+n]
    const float* p = vbase + (size_t)kl * DH + nt + ln;
    bf.x = p[0];
    bf.y = p[DH];
    oacc = WMMA_F32(oacc, a, bf);
  }
  if (half == 1) {
#pragma unroll
    for (int r = 0; r < 8; ++r) avbuf[r + hb * 8][nt + ln] = oacc[r];
  }
  __syncthreads();
  if (half == 0) {
    const int h = bh & (Hh - 1);
#pragma unroll
    for (int r = 0; r < 8; ++r) {
      const int ql = r + hb * 8;
      const float v = oacc[r] + avbuf[ql][nt + ln];
      // merged-head layout [B,S,D] for the final projection
      oh[((size_t)b * Ss + q0 + ql) * Dd + h * DH + nt + ln] = v;
    }
  }
}

// ---------------------------------------------------------------------------
// Launcher. Workspace: qh | kh | vh | oh  (4 * 8 MB = 32 MB).
// d_out = out [B,S,D] (2,097,152 f32) followed by attn [B,H,S,S].
// ---------------------------------------------------------------------------
extern "C" void kernel_launch(void* const* d_in, const int* in_sizes, int n_in,
                              void* d_out, int out_size, void* d_ws, size_t ws_size,
                              hipStream_t stream) {
  const float* v    = (const float*)d_in[0];
  const float* k    = (const float*)d_in[1];
  const float* q    = (const float*)d_in[2];
  const float* mind = (const float*)d_in[3];
  const float* wq = (const float*)d_in[4];  const float* bq = (const float*)d_in[5];
  const float* wk = (const float*)d_in[6];  const float* bk = (const float*)d_in[7];
  const float* wv = (const float*)d_in[8];  const float* bv = (const float*)d_in[9];
  const float* wo = (const float*)d_in[10]; const float* bo = (const float*)d_in[11];
  const float* sw1 = (const float*)d_in[12]; const float* sb1 = (const float*)d_in[13];
  const float* sw2 = (const float*)d_in[14]; const float* sb2 = (const float*)d_in[15];

  float* out  = (float*)d_out;
  float* attn = out + (size_t)Bb * Ss * Dd;

  float* ws = (float*)d_ws;
  float* qh = ws;
  float* kh = ws + PROJ_ELEMS;
  float* vh = ws + 2 * PROJ_ELEMS;
  float* oh = ws + 3 * PROJ_ELEMS;

  const dim3 blk(256);
  const dim3 gproj(MROWS / 32, Dd / 64);   // 128 x 8

  proj_gemm<<<gproj, blk, 0, stream>>>(q, wq, bq, qh, 1);
  proj_gemm<<<gproj, blk, 0, stream>>>(k, wk, bk, kh, 1);
  proj_gemm<<<gproj, blk, 0, stream>>>(v, wv, bv, vh, 1);

  attention_kernel<<<dim3(Bb * Hh * (Ss / 16)), blk, 0, stream>>>(
      qh, kh, vh, mind, sw1, sb1, sw2, sb2, attn, oh);

  proj_gemm<<<gproj, blk, 0, stream>>>(oh, wo, bo, out, 0);
}